// SimpleSAGE_36747740184682
// MI455X (gfx1250) — compile-verified
//
#include <hip/hip_runtime.h>
#include <math.h>

#define IN_C 128
#define EMB  64

typedef __attribute__((ext_vector_type(2))) float v2f;
typedef __attribute__((ext_vector_type(4))) float v4f;
typedef __attribute__((ext_vector_type(8))) float v8f;

// ---------------- utility kernels ----------------

__global__ void zero_i32_kernel(int* __restrict__ p, int n) {
    int i = blockIdx.x * blockDim.x + threadIdx.x;
    if (i < n) p[i] = 0;
}

__global__ void copy_i32_kernel(const int* __restrict__ a, int* __restrict__ b, int n) {
    int i = blockIdx.x * blockDim.x + threadIdx.x;
    if (i < n) b[i] = a[i];
}

// in-degree histogram
__global__ void hist_kernel(const int* __restrict__ dst, int E, int* __restrict__ cnt) {
    int e = blockIdx.x * blockDim.x + threadIdx.x;
    if (e < E) atomicAdd(&cnt[dst[e]], 1);
}

// single-block exclusive scan over n counts -> row_start[0..n]
__global__ void scan_kernel(const int* __restrict__ cnt, int* __restrict__ row_start, int n) {
    __shared__ int sm[256];
    __shared__ int carry;
    if (threadIdx.x == 0) carry = 0;
    __syncthreads();
    for (int base = 0; base < n; base += 256) {
        int i = base + (int)threadIdx.x;
        int v = (i < n) ? cnt[i] : 0;
        sm[threadIdx.x] = v;
        __syncthreads();
        // Hillis-Steele inclusive scan in LDS
        for (int off = 1; off < 256; off <<= 1) {
            int t = (threadIdx.x >= (unsigned)off) ? sm[threadIdx.x - off] : 0;
            __syncthreads();
            sm[threadIdx.x] += t;
            __syncthreads();
        }
        if (i < n) row_start[i] = carry + sm[threadIdx.x] - v;  // exclusive
        __syncthreads();
        if (threadIdx.x == 0) carry += sm[255];
        __syncthreads();
    }
    if (threadIdx.x == 0) row_start[n] = carry;
}

// bucket edge sources by destination (CSR fill)
__global__ void fill_kernel(const int* __restrict__ src, const int* __restrict__ dst, int E,
                            int* __restrict__ cursor, int* __restrict__ csr_src) {
    int e = blockIdx.x * blockDim.x + threadIdx.x;
    if (e < E) {
        int d   = dst[e];
        int pos = atomicAdd(&cursor[d], 1);
        csr_src[pos] = src[e];
    }
}

// ---------------- mean aggregation (wave-per-node, C = 128) ----------------

__global__ void aggregate_kernel(const float* __restrict__ X,
                                 const int* __restrict__ row_start,
                                 const int* __restrict__ csr_src,
                                 float* __restrict__ AGG, int N) {
    int wave = (int)((blockIdx.x * blockDim.x + threadIdx.x) >> 5);
    int lane = (int)(threadIdx.x & 31);
    if (wave >= N) return;
    int beg = row_start[wave];
    int end = row_start[wave + 1];
    const v4f* xv = (const v4f*)X;           // 32 x v4f per row of 128 floats
    v4f acc = (v4f){0.0f, 0.0f, 0.0f, 0.0f};
    for (int j = beg; j < end; ++j) {
        int s = csr_src[j];                  // wave-uniform -> scalar load
        acc += xv[(size_t)s * 32 + lane];
    }
    float invd = 1.0f / fmaxf((float)(end - beg), 1.0f);
    ((v4f*)AGG)[(size_t)wave * 32 + lane] = acc * invd;
}

// ---------------- fused SAGE GEMM via V_WMMA_F32_16X16X4_F32 ----------------
// Out[M, 16*NT] = act( AGG @ Wl^T + X @ Wr^T + bias ), C_in = 128 per operand.
// One wave computes a 16-row stripe across all NT 16-col tiles.

template<int NT, bool RELU>
__global__ void sage_gemm_kernel(const float* __restrict__ AGG, const float* __restrict__ X,
                                 const float* __restrict__ Wl,  const float* __restrict__ Wr,
                                 const float* __restrict__ bias, float* __restrict__ Out,
                                 int Mtiles, int N) {
    constexpr int C  = 128;
    constexpr int Nc = NT * 16;
    int wave = (int)((blockIdx.x * blockDim.x + threadIdx.x) >> 5);
    if (wave >= Mtiles) return;
    int lane = (int)(threadIdx.x & 31);
    int half = lane >> 4;        // lanes 16-31 carry K+2 (A) / K+2 rows (B)
    int lr   = lane & 15;        // M index for A, N index for B/C/D

    int m = wave * 16 + lr;
    if (m >= N) m = N - 1;       // safe duplicate read (stores are guarded)
    const float* arow = AGG + (size_t)m * C;
    const float* xrow = X   + (size_t)m * C;

    v8f acc[NT];
#pragma unroll
    for (int t = 0; t < NT; ++t) {
        float bb = bias[t * 16 + lr];        // bias depends only on column
#pragma unroll
        for (int i = 0; i < 8; ++i) acc[t][i] = bb;
    }

    // virtual K = 2*C : [mean-agg | x] against [Wl | Wr]
    for (int k0 = 0; k0 < 2 * C; k0 += 4) {
        int kk = k0 + half * 2;
        const bool left = (kk < C);
        const float* ap = left ? (arow + kk) : (xrow + (kk - C));
        v2f a;
        a.x = ap[0];
        a.y = ap[1];
        const float* W = left ? Wl : Wr;
        int kw = left ? kk : (kk - C);
#pragma unroll
        for (int t = 0; t < NT; ++t) {
            const float* wrow = W + (size_t)(t * 16 + lr) * C + kw;
            v2f b;
            b.x = wrow[0];                    // B[k][n]   = Wcat[n][k]
            b.y = wrow[1];                    // B[k+1][n] = Wcat[n][k+1]
            acc[t] = __builtin_amdgcn_wmma_f32_16x16x4_f32(
                false, a, false, b, (short)0, acc[t], false, false);
        }
    }

#pragma unroll
    for (int t = 0; t < NT; ++t) {
#pragma unroll
        for (int i = 0; i < 8; ++i) {
            float v = acc[t][i];
            if (RELU) v = fmaxf(v, 0.0f);
            int row = wave * 16 + half * 8 + i;   // C/D layout: lanes16-31 -> M+8
            if (row < N) Out[(size_t)row * Nc + t * 16 + lr] = v;
        }
    }
}

// ---------------- row-wise L2 normalize (wave-per-node, 64 cols) ----------------

__global__ void l2norm_kernel(float* __restrict__ out, int N) {
    int wave = (int)((blockIdx.x * blockDim.x + threadIdx.x) >> 5);
    int lane = (int)(threadIdx.x & 31);
    if (wave >= N) return;
    float2* p = (float2*)(out + (size_t)wave * EMB);
    float2 v = p[lane];
    float s = v.x * v.x + v.y * v.y;
#pragma unroll
    for (int off = 16; off > 0; off >>= 1) s += __shfl_xor(s, off, 32);
    float scale = 1.0f / fmaxf(sqrtf(s), 1e-12f);
    v.x *= scale;
    v.y *= scale;
    p[lane] = v;
}

// ---------------- host-side orchestration ----------------

extern "C" void kernel_launch(void* const* d_in, const int* in_sizes, int n_in,
                              void* d_out, int out_size, void* d_ws, size_t ws_size,
                              hipStream_t stream) {
    (void)n_in; (void)out_size; (void)ws_size;

    const float* x    = (const float*)d_in[0];
    const int*   edges= (const int*)d_in[1];     // int64 request -> int32 under default jax x64-off
    const float* W1l  = (const float*)d_in[2];
    const float* W1r  = (const float*)d_in[3];
    const float* b1   = (const float*)d_in[4];
    const float* W2l  = (const float*)d_in[5];
    const float* W2r  = (const float*)d_in[6];
    const float* b2   = (const float*)d_in[7];
    const float* W3l  = (const float*)d_in[8];
    const float* W3r  = (const float*)d_in[9];
    const float* b3   = (const float*)d_in[10];
    float* out = (float*)d_out;

    const int N = in_sizes[0] / IN_C;
    const int E = in_sizes[1] / 2;
    const int* src = edges;
    const int* dst = edges + E;

    // workspace carve-up (256B aligned)
    char* ws = (char*)d_ws;
    auto carve = [&](size_t bytes) {
        void* p = (void*)ws;
        ws += (bytes + 255) & ~(size_t)255;
        return p;
    };
    int*   cnt       = (int*)  carve((size_t)N * 4);
    int*   row_start = (int*)  carve((size_t)(N + 1) * 4);
    int*   cursor    = (int*)  carve((size_t)N * 4);
    int*   csr_src   = (int*)  carve((size_t)E * 4);
    float* agg       = (float*)carve((size_t)N * IN_C * 4);
    float* h1        = (float*)carve((size_t)N * IN_C * 4);
    float* h2        = (float*)carve((size_t)N * IN_C * 4);

    const dim3 blk(256);
    const int nb_N    = (N + 255) / 256;
    const int nb_E    = (E + 255) / 256;
    const int wpb     = 256 / 32;                     // 8 waves per block (wave32)
    const int nb_wavN = (N + wpb - 1) / wpb;          // wave-per-node kernels
    const int Mtiles  = (N + 15) / 16;
    const int nb_gemm = (Mtiles + wpb - 1) / wpb;

    // --- CSR build (once per call, deterministic) ---
    zero_i32_kernel<<<nb_N, blk, 0, stream>>>(cnt, N);
    hist_kernel    <<<nb_E, blk, 0, stream>>>(dst, E, cnt);
    scan_kernel    <<<1,    blk, 0, stream>>>(cnt, row_start, N);
    copy_i32_kernel<<<nb_N, blk, 0, stream>>>(row_start, cursor, N);
    fill_kernel    <<<nb_E, blk, 0, stream>>>(src, dst, E, cursor, csr_src);

    // --- layer 1: h1 = relu(mean(x) @ W1l^T + x @ W1r^T + b1) ---
    aggregate_kernel<<<nb_wavN, blk, 0, stream>>>(x, row_start, csr_src, agg, N);
    sage_gemm_kernel<8, true><<<nb_gemm, blk, 0, stream>>>(agg, x, W1l, W1r, b1, h1, Mtiles, N);

    // --- layer 2: h2 = relu(mean(h1) @ W2l^T + h1 @ W2r^T + b2) ---
    aggregate_kernel<<<nb_wavN, blk, 0, stream>>>(h1, row_start, csr_src, agg, N);
    sage_gemm_kernel<8, true><<<nb_gemm, blk, 0, stream>>>(agg, h1, W2l, W2r, b2, h2, Mtiles, N);

    // --- layer 3: out = mean(h2) @ W3l^T + h2 @ W3r^T + b3 (no relu) ---
    aggregate_kernel<<<nb_wavN, blk, 0, stream>>>(h2, row_start, csr_src, agg, N);
    sage_gemm_kernel<4, false><<<nb_gemm, blk, 0, stream>>>(agg, h2, W3l, W3r, b3, out, Mtiles, N);

    // --- row-wise L2 normalize in place ---
    l2norm_kernel<<<nb_wavN, blk, 0, stream>>>(out, N);
}